// TropicalConv2D_9749575762016
// MI455X (gfx1250) — compile-verified
//
#include <hip/hip_runtime.h>

#define AS1 __attribute__((address_space(1)))
#define AS3 __attribute__((address_space(3)))

typedef int v4i __attribute__((vector_size(4 * sizeof(int))));

#if __has_builtin(__builtin_amdgcn_global_load_async_to_lds_b128) && \
    __has_builtin(__builtin_amdgcn_s_wait_asynccnt)
#define HAVE_ASYNC 1
#else
#define HAVE_ASYNC 0
#endif

namespace {
constexpr int KHW   = 3;
constexpr int C     = 64;                 // input channels
constexpr int F     = 64;                 // output channels
constexpr int H     = 32, W = 32;
constexpr int HO    = 30, WO = 30;
constexpr int NPIX  = 4 * HO * WO;        // 3600 output pixels
constexpr int TAPS  = KHW * KHW;          // 9
constexpr int TAPF  = C * F;              // 4096 floats per kernel tap-chunk (16 KB)
constexpr int THREADS        = 256;       // 8 waves (wave32)
constexpr int PIX_PER_WAVE   = 2;
constexpr int WAVES          = THREADS / 32;
constexpr int PIX_PER_BLOCK  = WAVES * PIX_PER_WAVE;  // 16
}

// Stage one 64x64 f32 tap-chunk of the kernel matrix into LDS.
// Async path: 4 x global_load_async_to_lds_b128 per thread (tracked by ASYNCcnt).
__device__ __forceinline__ void stage_tap(const float* __restrict__ src,
                                          float* dst_lds, int tid) {
#if HAVE_ASYNC
#pragma unroll
  for (int s = 0; s < 4; ++s) {
    const int e = (s * THREADS + tid) * 4;
    __builtin_amdgcn_global_load_async_to_lds_b128(
        (AS1 v4i*)(src + e), (AS3 v4i*)((AS3 float*)dst_lds + e),
        /*offset=*/0, /*cpol=*/0);
  }
#else
#pragma unroll
  for (int s = 0; s < 4; ++s) {
    const int e = (s * THREADS + tid) * 4;
    *(float4*)(dst_lds + e) = *(const float4*)(src + e);
  }
#endif
}

#if HAVE_ASYNC
#define ASYNC_WAIT(n) __builtin_amdgcn_s_wait_asynccnt(n)
#else
#define ASYNC_WAIT(n)
#endif

__global__ __launch_bounds__(THREADS)
void tropical_conv2d_9749575762016_kernel(const float* __restrict__ x,
                                          const float* __restrict__ kern,
                                          float* __restrict__ out) {
  __shared__ float ldsK[2][TAPF];   // 32 KB double buffer of kernel tap-chunks

  const int tid  = threadIdx.x;
  const int lane = tid & 31;
  // Wave id; readfirstlane pins it to an SGPR so all per-pixel x addressing
  // (uniform within the wave) scalarizes to SMEM loads.
  const int wv = __builtin_amdgcn_readfirstlane(tid >> 5);

  // ---- prologue: stage tap 0 into buffer 0 ----
  stage_tap(kern, &ldsK[0][0], tid);

  // ---- wave-uniform pixel bookkeeping (2 pixels per wave) ----
  const int pix0 = blockIdx.x * PIX_PER_BLOCK + wv * PIX_PER_WAVE;
  const int pix1 = pix0 + 1;                       // NPIX even -> always valid
  const int n0 = pix0 / (HO * WO);
  const int r0 = pix0 - n0 * (HO * WO);
  const int ho0 = r0 / WO, wo0 = r0 - (r0 / WO) * WO;
  const int n1 = pix1 / (HO * WO);
  const int r1 = pix1 - n1 * (HO * WO);
  const int ho1 = r1 / WO, wo1 = r1 - (r1 / WO) * WO;
  const int xbase0 = ((n0 * H + ho0) * W + wo0) * C;
  const int xbase1 = ((n1 * H + ho1) * W + wo1) * C;

  // accumulators: 2 pixels x 2 channels (f = 2*lane, 2*lane+1)
  float acc00 = -__builtin_inff();
  float acc01 = -__builtin_inff();
  float acc10 = -__builtin_inff();
  float acc11 = -__builtin_inff();

  for (int t = 0; t < TAPS; ++t) {
    // Prefetch next tap into the other buffer, then wait until the CURRENT
    // tap's async loads (issued one iteration ago) have landed. Async loads
    // complete in order, so ASYNCcnt <= 4 means tap t is resident.
    if (t + 1 < TAPS) {
      stage_tap(kern + (t + 1) * TAPF, &ldsK[(t + 1) & 1][0], tid);
      ASYNC_WAIT(4);
    } else {
      ASYNC_WAIT(0);
    }
    __syncthreads();

    const int di = t / KHW;
    const int dj = t - di * KHW;
    const float* __restrict__ xs0 = x + xbase0 + (di * W + dj) * C;  // uniform
    const float* __restrict__ xs1 = x + xbase1 + (di * W + dj) * C;  // uniform
    const float* kb = &ldsK[t & 1][0];

#pragma unroll 8
    for (int c = 0; c < C; ++c) {
      const float  xv0 = xs0[c];                       // wave-uniform (SMEM)
      const float  xv1 = xs1[c];                       // wave-uniform (SMEM)
      const float2 kv  = *(const float2*)&kb[c * F + 2 * lane];  // ds_load_b64
      acc00 = fmaxf(acc00, xv0 + kv.x);
      acc01 = fmaxf(acc01, xv0 + kv.y);
      acc10 = fmaxf(acc10, xv1 + kv.x);
      acc11 = fmaxf(acc11, xv1 + kv.y);
    }
    __syncthreads();  // all waves done reading ldsK[t&1] before it is refilled
  }

  // BETA == 1.0 -> identity scale; coalesced float2 stores per pixel.
  float2 o0; o0.x = acc00; o0.y = acc01;
  float2 o1; o1.x = acc10; o1.y = acc11;
  *(float2*)&out[pix0 * F + 2 * lane] = o0;
  *(float2*)&out[pix1 * F + 2 * lane] = o1;
}

extern "C" void kernel_launch(void* const* d_in, const int* in_sizes, int n_in,
                              void* d_out, int out_size, void* d_ws, size_t ws_size,
                              hipStream_t stream) {
  const float* x    = (const float*)d_in[0];   // (4,32,32,64) f32
  const float* kern = (const float*)d_in[1];   // (576,64)     f32
  float* out        = (float*)d_out;           // (4,30,30,64) f32
  (void)in_sizes; (void)n_in; (void)out_size; (void)d_ws; (void)ws_size;

  dim3 grid(NPIX / PIX_PER_BLOCK);  // 225 blocks
  dim3 block(THREADS);              // 8 wave32 waves
  tropical_conv2d_9749575762016_kernel<<<grid, block, 0, stream>>>(x, kern, out);
}